// TransductiveMDGCNLayer_773094113325
// MI455X (gfx1250) — compile-verified
//
#include <hip/hip_runtime.h>

typedef __attribute__((ext_vector_type(2))) float v2f;
typedef __attribute__((ext_vector_type(8))) float v8f;

#define N_NODES 100000
#define DIM     128
#define NHOPS   3
#define NEDGE   3200000
#define EDIM    10
#define WSTRIDE 132   // transposed-W LDS row stride: 528B (8B aligned),
                      // bank(n,k) = (4n + k) % 64 -> conflict-free b64 reads

// ---------------------------------------------------------------------------
// Kernel 1: xt = X @ W   via V_WMMA_F32_16X16X4_F32 (full fp32, matches ref)
// Block = 128 threads = 4 waves. W staged TRANSPOSED in LDS once per block:
// Ws[n*WSTRIDE + k], so a B fragment {W[kk][n], W[kk+1][n]} is ONE aligned
// ds_load_b64 into a register pair (no VGPR re-packing before the WMMA).
// Each wave computes two 16-row panels (block covers 128 rows) to amortize
// the staging. 8 independent 16x16 C tiles per panel pipeline the WMMAs.
// N % 16 == 0 so a wave's 16-row panel is either fully valid or skipped.
// ---------------------------------------------------------------------------
__global__ __launch_bounds__(128) void gemm_xw_wmma(
    const float* __restrict__ X, const float* __restrict__ W,
    float* __restrict__ XT) {
  __shared__ float Ws[DIM * WSTRIDE];  // 67.6 KB of the 320 KB WGP LDS

  // Stage W transposed: thread i handles k-pair kp = i>>7, column n = i&127.
  // Global reads: consecutive lanes read consecutive n (coalesced rows).
  // LDS writes: float2 at Ws[n*WSTRIDE + 2kp] (8B aligned, <=2-way conflict).
  for (int i = threadIdx.x; i < (DIM / 2) * DIM; i += blockDim.x) {
    const int kp = i >> 7;
    const int n  = i & 127;
    float2 w2;
    w2.x = W[(2 * kp + 0) * DIM + n];
    w2.y = W[(2 * kp + 1) * DIM + n];
    *(float2*)(&Ws[n * WSTRIDE + 2 * kp]) = w2;
  }
  __syncthreads();

  const int wave  = threadIdx.x >> 5;
  const int lane  = threadIdx.x & 31;
  const int m     = lane & 15;        // row within 16-tile (A) / col (B,C)
  const int halfk = (lane >> 4) * 2;  // K sub-pair selected by lane half
  const int rowoff = (lane >> 4) * 8; // C/D: VGPR v -> row v + 8*(lane>>4)

#pragma unroll
  for (int p = 0; p < 2; ++p) {
    const int r0 = blockIdx.x * 128 + p * 64 + wave * 16;
    if (r0 >= N_NODES) continue;

    v8f c[8] = {};  // 8 column tiles x 8 VGPRs accumulators
    const float* xrow = X + (size_t)(r0 + m) * DIM;

    for (int k0 = 0; k0 < DIM; k0 += 4) {
      const int kk = k0 + halfk;
      // A fragment: 16x4 fp32; lanes 0-15 K={k0,k0+1}, lanes 16-31 K={k0+2,k0+3}
      v2f a = *(const v2f*)(xrow + kk);
#pragma unroll
      for (int t = 0; t < 8; ++t) {
        // B fragment: single aligned ds_load_b64 from transposed W
        v2f b = *(const v2f*)(&Ws[(t * 16 + m) * WSTRIDE + kk]);
        c[t] = __builtin_amdgcn_wmma_f32_16x16x4_f32(
            /*neg_a=*/false, a, /*neg_b=*/false, b,
            /*c_mod=*/(short)0, c[t], /*reuse_a=*/false, /*reuse_b=*/false);
      }
    }

#pragma unroll
    for (int t = 0; t < 8; ++t)
#pragma unroll
      for (int v = 0; v < 8; ++v)
        XT[(size_t)(r0 + rowoff + v) * DIM + t * 16 + m] = c[t][v];
  }
}

// ---------------------------------------------------------------------------
// Kernel 2: COO scatter  out[row] += val * xt[col]   (segment-sum via atomics)
// One edge per 32-lane group; float4 per lane covers the 128-wide row.
// xt and out both fit in the 192MB L2, so gathers/atomics are L2-resident.
// ---------------------------------------------------------------------------
__global__ __launch_bounds__(256) void spmm_scatter(
    const int* __restrict__ rows, const int* __restrict__ cols,
    const float* __restrict__ vals, const float* __restrict__ XT,
    float* __restrict__ OUT) {
  const long long idx  = (long long)blockIdx.x * blockDim.x + threadIdx.x;
  const int       edge = (int)(idx >> 5);
  const int       lane = (int)(idx & 31);
  if (edge >= NEDGE) return;

  const int   cIdx = cols[edge];
  const int   rIdx = rows[edge];
  const float v    = vals[edge];

  // gfx1250 prefetch of an upcoming gathered row (global_prefetch_b8)
  if (lane == 0 && edge + 64 < NEDGE) {
    const int cn = cols[edge + 64];
    __builtin_prefetch(XT + (size_t)cn * DIM, 0, 0);
  }

  const float4 x = *(const float4*)(XT + (size_t)cIdx * DIM + lane * 4);
  float*       o = OUT + (size_t)rIdx * DIM + lane * 4;
  atomicAdd(o + 0, v * x.x);  // global_atomic_add_f32, no-return (STOREcnt)
  atomicAdd(o + 1, v * x.y);
  atomicAdd(o + 2, v * x.z);
  atomicAdd(o + 3, v * x.w);
}

// ---------------------------------------------------------------------------
// Kernel 3: Psum += embed2^T @ xt  (10 x 128 reduction over N nodes).
// Thread d owns column d; 10-float register accumulator; a few global atomics
// per block at the end (256 blocks x 1280 floats -> negligible contention).
// ---------------------------------------------------------------------------
__global__ __launch_bounds__(128) void lowrank_reduce(
    const float* __restrict__ XT, const float* __restrict__ E2,
    float* __restrict__ Psum) {
  const int d = threadIdx.x;
  float acc[EDIM];
#pragma unroll
  for (int e = 0; e < EDIM; ++e) acc[e] = 0.f;

  for (int i = blockIdx.x; i < N_NODES; i += gridDim.x) {
    const float x = XT[(size_t)i * DIM + d];
#pragma unroll
    for (int e = 0; e < EDIM; ++e) acc[e] += E2[i * EDIM + e] * x;
  }
#pragma unroll
  for (int e = 0; e < EDIM; ++e) atomicAdd(&Psum[e * DIM + d], acc[e]);
}

// ---------------------------------------------------------------------------
// Kernel 4: out = relu(out + alpha * embed1 @ Psum)
// Psum (1280 floats) cached in LDS; thread d owns column d, 32 rows per block.
// ---------------------------------------------------------------------------
__global__ __launch_bounds__(128) void finalize_relu(
    float* __restrict__ OUT, const float* __restrict__ E1,
    const float* __restrict__ Psum, const float* __restrict__ alpha_p) {
  __shared__ float Ps[EDIM * DIM];
  for (int i = threadIdx.x; i < EDIM * DIM; i += blockDim.x) Ps[i] = Psum[i];
  __syncthreads();

  const float a = *alpha_p;
  const int   d = threadIdx.x;
  const int   rEnd =
      (blockIdx.x * 32 + 32 < N_NODES) ? blockIdx.x * 32 + 32 : N_NODES;
  for (int r = blockIdx.x * 32; r < rEnd; ++r) {
    float s = 0.f;
#pragma unroll
    for (int e = 0; e < EDIM; ++e) s += E1[r * EDIM + e] * Ps[e * DIM + d];
    const float o = OUT[(size_t)r * DIM + d] + a * s;
    OUT[(size_t)r * DIM + d] = o > 0.f ? o : 0.f;
  }
}

// ---------------------------------------------------------------------------
extern "C" void kernel_launch(void* const* d_in, const int* in_sizes, int n_in,
                              void* d_out, int out_size, void* d_ws,
                              size_t ws_size, hipStream_t stream) {
  const float* X     = (const float*)d_in[0];  // [N,128]
  const int*   erows = (const int*)d_in[1];    // [3,E]
  const int*   ecols = (const int*)d_in[2];    // [3,E]
  const float* evals = (const float*)d_in[3];  // [3,E]
  const float* W     = (const float*)d_in[4];  // [3,128,128]
  const float* E1    = (const float*)d_in[5];  // [N,10]
  const float* E2    = (const float*)d_in[6];  // [N,10]
  const float* alpha = (const float*)d_in[7];  // scalar

  float* OUT  = (float*)d_out;
  float* XT   = (float*)d_ws;                  // 51.2 MB scratch
  float* Psum = XT + (size_t)N_NODES * DIM;    // 1280 floats

  hipMemsetAsync(OUT, 0, (size_t)N_NODES * DIM * sizeof(float), stream);
  hipMemsetAsync(Psum, 0, (size_t)EDIM * DIM * sizeof(float), stream);

  const int gemmBlocks    = (N_NODES + 127) / 128;
  const int scatterBlocks = (int)(((long long)NEDGE * 32 + 255) / 256);

  for (int hop = 0; hop < NHOPS; ++hop) {
    gemm_xw_wmma<<<gemmBlocks, 128, 0, stream>>>(
        X, W + (size_t)hop * DIM * DIM, XT);
    spmm_scatter<<<scatterBlocks, 256, 0, stream>>>(
        erows + (size_t)hop * NEDGE, ecols + (size_t)hop * NEDGE,
        evals + (size_t)hop * NEDGE, XT, OUT);
    lowrank_reduce<<<256, 128, 0, stream>>>(XT, E2, Psum);
  }
  finalize_relu<<<(N_NODES + 31) / 32, 128, 0, stream>>>(OUT, E1, Psum, alpha);
}